// Basis_Change_Image_to_larger_Fock_density_37709812859571
// MI455X (gfx1250) — compile-verified
//
#include <hip/hip_runtime.h>
#include <math.h>

typedef float v4f __attribute__((ext_vector_type(4)));
typedef int   v4i __attribute__((ext_vector_type(4)));

#define FD_MAX 8192  // 32 KB LDS for the inverse table (fock_dim = 4656 fits)

// CDNA5 async global->LDS path, guarded so compilation never breaks.
#if defined(__AMDGCN__) && __has_builtin(__builtin_amdgcn_global_load_async_to_lds_b32)
#define HAVE_ASYNC_LDS 1
#else
#define HAVE_ASYNC_LDS 0
#endif

__global__ __launch_bounds__(256) void bc_init_inv(int* __restrict__ inv, int fd) {
  int i = blockIdx.x * 256 + threadIdx.x;
  if (i < fd) inv[i] = -1;
}

__global__ __launch_bounds__(256) void bc_fill_inv(const int* __restrict__ fidx,
                                                   int* __restrict__ inv, int n) {
  int c = blockIdx.x * 256 + threadIdx.x;
  if (c < n) inv[fidx[c]] = c;  // map is injective -> no races
}

__device__ __forceinline__ void stage_inv_to_lds(const int* __restrict__ inv,
                                                 int* s_inv, int fd) {
#if HAVE_ASYNC_LDS
  for (int i = threadIdx.x; i < fd; i += blockDim.x) {
    __builtin_amdgcn_global_load_async_to_lds_b32(
        (__attribute__((address_space(1))) int*)(inv + i),
        (__attribute__((address_space(3))) int*)(s_inv + i),
        /*offset=*/0, /*cpol=*/0);
  }
#if __has_builtin(__builtin_amdgcn_s_wait_asynccnt)
  __builtin_amdgcn_s_wait_asynccnt(0);
#else
  asm volatile("s_wait_asynccnt 0" ::: "memory");
#endif
#else
  for (int i = threadIdx.x; i < fd; i += blockDim.x) s_inv[i] = inv[i];
#endif
  __syncthreads();
}

// Single streaming pass over the output: each row is either all-zero
// (NT b128 streaming stores) or a gather of one 4KB input row permuted by inv.
__global__ __launch_bounds__(256) void bc_scatter_vec4(
    const float* __restrict__ in, float* __restrict__ out,
    const int* __restrict__ inv, int fd, int n, int totalRows) {
  __shared__ __align__(16) int s_inv[FD_MAX];
  stage_inv_to_lds(inv, s_inv, fd);

  const int fd4 = fd >> 2;
  for (int row = blockIdx.x; row < totalRows; row += gridDim.x) {
    const int b = row / fd;
    const int r = row - b * fd;
    v4f* __restrict__ orow = (v4f*)(out + (size_t)row * (size_t)fd);
    const int ri = s_inv[r];
    if (ri < 0) {
      // zero row: pure streaming NT b128 stores
      const v4f z = {0.0f, 0.0f, 0.0f, 0.0f};
      for (int c4 = threadIdx.x; c4 < fd4; c4 += blockDim.x)
        __builtin_nontemporal_store(z, orow + c4);
    } else {
      const float* __restrict__ irow =
          in + ((size_t)b * (size_t)n + (size_t)ri) * (size_t)n;
      // prefetch the 4KB input row (global_prefetch_b8) to hide gather latency
      const int lines = (n * 4 + 63) >> 6;
      for (int p = threadIdx.x; p < lines; p += blockDim.x)
        __builtin_prefetch((const char*)irow + (p << 6), 0, 1);
      for (int c4 = threadIdx.x; c4 < fd4; c4 += blockDim.x) {
        const v4i ci = *(const v4i*)(s_inv + (c4 << 2));  // ds_load_b128
        v4f v;
        v.x = (ci.x >= 0) ? irow[ci.x] : 0.0f;
        v.y = (ci.y >= 0) ? irow[ci.y] : 0.0f;
        v.z = (ci.z >= 0) ? irow[ci.z] : 0.0f;
        v.w = (ci.w >= 0) ? irow[ci.w] : 0.0f;
        __builtin_nontemporal_store(v, orow + c4);
      }
    }
  }
}

// Fallback for fd not divisible by 4 or fd > FD_MAX (not hit for this problem).
__global__ __launch_bounds__(256) void bc_scatter_scalar(
    const float* __restrict__ in, float* __restrict__ out,
    const int* __restrict__ inv, int fd, int n, int totalRows) {
  for (int row = blockIdx.x; row < totalRows; row += gridDim.x) {
    const int b = row / fd;
    const int r = row - b * fd;
    float* __restrict__ orow = out + (size_t)row * (size_t)fd;
    const int ri = inv[r];
    if (ri < 0) {
      for (int c = threadIdx.x; c < fd; c += blockDim.x)
        __builtin_nontemporal_store(0.0f, orow + c);
    } else {
      const float* __restrict__ irow =
          in + ((size_t)b * (size_t)n + (size_t)ri) * (size_t)n;
      for (int c = threadIdx.x; c < fd; c += blockDim.x) {
        const int ci = inv[c];
        __builtin_nontemporal_store(ci >= 0 ? irow[ci] : 0.0f, orow + c);
      }
    }
  }
}

extern "C" void kernel_launch(void* const* d_in, const int* in_sizes, int n_in,
                              void* d_out, int out_size, void* d_ws, size_t ws_size,
                              hipStream_t stream) {
  const float* in   = (const float*)d_in[0];  // [B, n, n] fp32
  const int*   fidx = (const int*)d_in[1];    // [n] int32
  float* out = (float*)d_out;                 // [B, fd, fd] fp32

  const int n = in_sizes[1];                  // 1024
  const long long nn = (long long)n * (long long)n;
  const int B = (int)((long long)in_sizes[0] / nn);          // 8
  const long long per = (long long)out_size / (long long)B;  // fd*fd
  const int fd = (int)(sqrt((double)per) + 0.5);             // 4656

  int* inv = (int*)d_ws;  // fd ints of workspace

  bc_init_inv<<<(fd + 255) / 256, 256, 0, stream>>>(inv, fd);
  bc_fill_inv<<<(n + 255) / 256, 256, 0, stream>>>(fidx, inv, n);

  const int totalRows = B * fd;  // 37248
  int nb = totalRows < 2048 ? totalRows : 2048;
  if ((fd % 4 == 0) && fd <= FD_MAX) {
    bc_scatter_vec4<<<nb, 256, 0, stream>>>(in, out, inv, fd, n, totalRows);
  } else {
    bc_scatter_scalar<<<nb, 256, 0, stream>>>(in, out, inv, fd, n, totalRows);
  }
}